// MVDNet_prop_61632780698034
// MI455X (gfx1250) — compile-verified
//
#include <hip/hip_runtime.h>
#include <hip/hip_bf16.h>

// ---------------------------------------------------------------------------
// MVDNet propagation: 3x3 conv (WMMA f32, im2col-in-LDS) -> deformable
// gathers -> affinity. Shapes per reference: B=4, Cg=16, H=W=384, num=Cf=8.
// ---------------------------------------------------------------------------

typedef float v2f __attribute__((ext_vector_type(2)));
typedef float v8f __attribute__((ext_vector_type(8)));

#define BATCH 4
#define CG    16
#define HH    384
#define WW    384
#define HW    (HH * WW)
#define NUM   8
#define KTOT  144           // 16 ci * 3 * 3
#define OCH   24            // 3 * NUM
#define KPAD  146           // KTOT padded: even (b64 align) and 146%64=18 -> spread banks

// ---------------------------------------------------------------------------
// Kernel 1: 3x3 SAME conv via V_WMMA_F32_16X16X4_F32.
// GEMM: out[pix, ch] = sum_k patch[pix, k] * Wt[k, ch],  k = ci*9 + ky*3 + kx.
// Block = 128 threads = 4 waves = one 64-pixel strip of one image row.
// Stage im2col tile tileA[c][k] and weights ldsw[ch][k] in LDS, then the hot
// loop is 3x ds_load_b64 + 2x v_wmma per K-step, fully unrolled, no branches.
// ---------------------------------------------------------------------------
__global__ __launch_bounds__(128) void conv3x3_wmma_kernel(
    const float* __restrict__ g,       // (B, 16, H, W)
    const float* __restrict__ Woa,     // (24, 16, 3, 3) -> Wt[k, ch] = Woa[ch*144 + k]
    const float* __restrict__ boa,     // (24,)
    float*       __restrict__ convout) // (B, 24, H, W)
{
    __shared__ float tileA[64 * KPAD];  // [pixel c][k]  im2col, pre-shifted
    __shared__ float ldsw [32 * KPAD];  // [ch][k]       ch padded 24->32 with 0

    const int tid = threadIdx.x;
    const int bid = blockIdx.x;                  // B * H * (W/64) blocks
    const int xg  = bid % (WW / 64);
    const int t   = bid / (WW / 64);
    const int y   = t % HH;
    const int b   = t / HH;
    const float* gb = g + (size_t)b * CG * HW;

    // ---- stage weights: ldsw[ch*KPAD + k] ----
    for (int idx = tid; idx < 32 * KTOT; idx += 128) {
        const int ch = idx / KTOT;
        const int k  = idx - ch * KTOT;
        ldsw[ch * KPAD + k] = (ch < OCH) ? Woa[ch * KTOT + k] : 0.0f;
    }

    // ---- stage im2col tile: tileA[c*KPAD + k] = patch(x0+c, k) ----
    // 128 threads = 2 groups x 64 cols; group handles 8 input channels.
    {
        const int c   = tid & 63;
        const int sub = tid >> 6;                // 0,1
        const int xcol = xg * 64 + c;            // global pixel column
        for (int cih = 0; cih < 8; ++cih) {
            const int ci = sub * 8 + cih;
            const float* gc = gb + (size_t)ci * HW;
#pragma unroll
            for (int r = 0; r < 9; ++r) {        // ky,kx constant-folded
                const int ky = r / 3, kx = r - ky * 3;
                const int yy = y + ky - 1;
                const int xx = xcol + kx - 1;
                float v = 0.0f;
                if (yy >= 0 && yy < HH && xx >= 0 && xx < WW)
                    v = gc[yy * WW + xx];
                tileA[c * KPAD + ci * 9 + r] = v;
            }
        }
    }
    __syncthreads();

    const int wave = tid >> 5;
    const int lane = tid & 31;
    const int x0   = xg * 64 + wave * 16;
    const int m    = lane & 15;                  // A-row (pixel) / B-col (ch)
    const int half = lane >> 4;                  // K half select (folded below)
    const int n    = m;

    // Lane bases: 2*half folds the A/B K-half layout into the base address.
    const float* aBase  = tileA + (wave * 16 + m) * KPAD + 2 * half;
    const float* b0Base = ldsw + n * KPAD + 2 * half;
    const float* b1Base = ldsw + (n + 16) * KPAD + 2 * half;

    v8f acc0 = {0.f,0.f,0.f,0.f,0.f,0.f,0.f,0.f};
    v8f acc1 = {0.f,0.f,0.f,0.f,0.f,0.f,0.f,0.f};

#pragma unroll
    for (int kk = 0; kk < KTOT / 4; ++kk) {
        const v2f a   = *reinterpret_cast<const v2f*>(aBase  + 4 * kk);
        const v2f b0v = *reinterpret_cast<const v2f*>(b0Base + 4 * kk);
        const v2f b1v = *reinterpret_cast<const v2f*>(b1Base + 4 * kk);
        acc0 = __builtin_amdgcn_wmma_f32_16x16x4_f32(
                   false, a, false, b0v, (short)0, acc0, false, false);
        acc1 = __builtin_amdgcn_wmma_f32_16x16x4_f32(
                   false, a, false, b1v, (short)0, acc1, false, false);
    }

    const float bias0 = boa[n];
    const float bias1 = (16 + n < OCH) ? boa[16 + n] : 0.0f;
    float* outb = convout + (size_t)b * OCH * HW;
#pragma unroll
    for (int r = 0; r < 8; ++r) {
        const int mp = half * 8 + r;             // D layout: M = half*8 + r
        const int xo = x0 + mp;
        outb[((size_t)n * HH + y) * WW + xo] = acc0[r] + bias0;
        if (16 + n < OCH)
            outb[((size_t)(16 + n) * HH + y) * WW + xo] = acc1[r] + bias1;
    }
}

// ---------------------------------------------------------------------------
// Helpers replicating the reference numerics exactly.
// ---------------------------------------------------------------------------
__device__ __forceinline__ float grid_roundtrip(float v) {
    // gx = 2v/(W-1) - 1 ; ix = (gx+1)*(W-1)/2   (H == W == 384)
    const float d = 383.0f;
    const float gx = 2.0f * v / d - 1.0f;
    return (gx + 1.0f) * d / 2.0f;
}

__device__ __forceinline__ float bilin384(const float* __restrict__ img,
                                          float iy, float ix) {
    const float y0  = floorf(iy), x0 = floorf(ix);
    const float wy1 = iy - y0,   wx1 = ix - x0;
    float acc = 0.0f;
#pragma unroll
    for (int dy = 0; dy < 2; ++dy) {
#pragma unroll
        for (int dx = 0; dx < 2; ++dx) {
            const float yc = y0 + (float)dy;
            const float xc = x0 + (float)dx;
            const bool ok = (yc >= 0.0f) && (yc <= 383.0f) &&
                            (xc >= 0.0f) && (xc <= 383.0f);
            const int yi = (int)fminf(fmaxf(yc, 0.0f), 383.0f);
            const int xi = (int)fminf(fmaxf(xc, 0.0f), 383.0f);
            const float wy = dy ? wy1 : (1.0f - wy1);
            const float wx = dx ? wx1 : (1.0f - wx1);
            acc += img[yi * WW + xi] * (wy * wx * (ok ? 1.0f : 0.0f));
        }
    }
    return acc;
}

// ---------------------------------------------------------------------------
// Kernel 2: per output pixel -> cos_w (reference's reshape-scramble solved in
// closed form), tanh/scale, confidence gather, normalization, softmax;
// writes offset_out (18 ch) and aff_full (9 ch).
//
// Scramble decode (H=W=384, C=num=8, 48 = H/8): for output (h',w',j,c'):
//   src channel c = h'/48 ; src pixel h = (h'%48)*8 + w'/48,
//   w = (w'%48)*8 + j ; tap n = c'.
// Feature-gather coords (reference quirk): ix = dy_n + (n<4 ? h : w),
//   iy = dx_n + (n<4 ? h : w), each passed through grid_roundtrip.
// ---------------------------------------------------------------------------
__global__ __launch_bounds__(256) void aff_kernel(
    const float* __restrict__ convo,   // ws: (B, 24, H, W)
    const float* __restrict__ fea,     // (B, 8, H, W)
    const float* __restrict__ conf,    // (B, 1, H, W)
    const float* __restrict__ scale_c, // (1,)
    float*       __restrict__ out_off, // (B, 18, H, W)
    float*       __restrict__ out_aff) // (B, 9, H, W)
{
    const int idx = blockIdx.x * 256 + threadIdx.x;
    if (idx >= BATCH * HW) return;
    const int wq = idx % WW;
    const int t  = idx / WW;
    const int hq = t % HH;
    const int b  = t / HH;

    const float* cb = convo + (size_t)b * OCH * HW;
    const float* fb = fea   + (size_t)b * NUM * HW;
    const float* kb = conf  + (size_t)b * HW;

    // ---- cos_w via the scrambled gather ----
    const int c  = hq / 48;
    const int hs = (hq % 48) * 8 + wq / 48;
    const int wb = (wq % 48) * 8;
    const float* fc = fb + (size_t)c * HW;

    const int p = hq * WW + wq;
    float feav[NUM];
#pragma unroll
    for (int j = 0; j < NUM; ++j) feav[j] = fb[(size_t)j * HW + p];

    float cosw[NUM];
#pragma unroll
    for (int nn = 0; nn < NUM; ++nn) cosw[nn] = 0.0f;

    for (int j = 0; j < NUM; ++j) {
        const int wsrc = wb + j;
        const int sp   = hs * WW + wsrc;
        const float fj = feav[j];
#pragma unroll
        for (int nn = 0; nn < NUM; ++nn) {
            const float dy   = cb[(size_t)nn * HW + sp];         // o1 channel nn
            const float dx   = cb[(size_t)(NUM + nn) * HW + sp]; // o2 channel nn
            const float base = (nn < 4) ? (float)hs : (float)wsrc;
            const float ix = grid_roundtrip(dy + base);
            const float iy = grid_roundtrip(dx + base);
            cosw[nn] += bilin384(fc, iy, ix) * fj;
        }
    }

    // ---- affinity chain at this pixel ----
    const float scale = scale_c[0] + 1e-8f;
    float dyo[NUM], dxo[NUM], av[NUM];
#pragma unroll
    for (int nn = 0; nn < NUM; ++nn) {
        dyo[nn] = cb[(size_t)nn * HW + p];
        dxo[nn] = cb[(size_t)(NUM + nn) * HW + p];
    }
#pragma unroll
    for (int nn = 0; nn < NUM; ++nn) {
        float a = cb[(size_t)(2 * NUM + nn) * HW + p] * cosw[nn];
        a = tanhf(a) / scale;
        const float cf = bilin384(kb, dyo[nn] + (float)hq, dxo[nn] + (float)wq);
        av[nn] = a * cf;
    }
    float s = 1e-4f;
#pragma unroll
    for (int nn = 0; nn < NUM; ++nn) s += fabsf(av[nn]);
    s = fmaxf(s, 1.0f);
    float ssum = 0.0f;
#pragma unroll
    for (int nn = 0; nn < NUM; ++nn) { av[nn] /= s; ssum += av[nn]; }

    float vec[NUM + 1];
#pragma unroll
    for (int nn = 0; nn < 4; ++nn) vec[nn] = av[nn];
    vec[4] = 1.0f - ssum;
#pragma unroll
    for (int nn = 4; nn < NUM; ++nn) vec[nn + 1] = av[nn];

    float mx = vec[0];
#pragma unroll
    for (int i = 1; i < 9; ++i) mx = fmaxf(mx, vec[i]);
    float es = 0.0f, ev[9];
#pragma unroll
    for (int i = 0; i < 9; ++i) { ev[i] = expf(vec[i] - mx); es += ev[i]; }

    float* ao = out_aff + (size_t)b * 9 * HW;
#pragma unroll
    for (int i = 0; i < 9; ++i) ao[(size_t)i * HW + p] = ev[i] / es;

    // ---- offset_out: [n<4: (dy,dx)], zero pair, [n>=4: (dy,dx)] ----
    float* oo = out_off + (size_t)b * 18 * HW;
#pragma unroll
    for (int nn = 0; nn < 4; ++nn) {
        oo[(size_t)(2 * nn)     * HW + p] = dyo[nn];
        oo[(size_t)(2 * nn + 1) * HW + p] = dxo[nn];
    }
    oo[(size_t)8 * HW + p] = 0.0f;
    oo[(size_t)9 * HW + p] = 0.0f;
#pragma unroll
    for (int nn = 4; nn < NUM; ++nn) {
        oo[(size_t)(2 * nn + 2) * HW + p] = dyo[nn];
        oo[(size_t)(2 * nn + 3) * HW + p] = dxo[nn];
    }
}

// ---------------------------------------------------------------------------
extern "C" void kernel_launch(void* const* d_in, const int* in_sizes, int n_in,
                              void* d_out, int out_size, void* d_ws, size_t ws_size,
                              hipStream_t stream) {
    const float* guidance = (const float*)d_in[0]; // (4,16,384,384)
    const float* gtconf   = (const float*)d_in[1]; // (4,1,384,384)
    const float* fea      = (const float*)d_in[2]; // (4,8,384,384)
    const float* Woa      = (const float*)d_in[3]; // (24,16,3,3)
    const float* boa      = (const float*)d_in[4]; // (24,)
    const float* scale    = (const float*)d_in[5]; // (1,)

    float* convws  = (float*)d_ws;                      // B*24*HW floats = 56.6 MB
    float* out_off = (float*)d_out;                     // (B,18,H,W)
    float* out_aff = out_off + (size_t)BATCH * 18 * HW; // (B,9,H,W)

    // K1: WMMA conv. One block = one 64-pixel row strip (4 waves x 16 px).
    const int nblk1 = BATCH * HH * (WW / 64); // 9216
    conv3x3_wmma_kernel<<<nblk1, 128, 0, stream>>>(guidance, Woa, boa, convws);

    // K2: gathers + affinity + outputs. One thread per output pixel.
    const int total = BATCH * HW;
    aff_kernel<<<(total + 255) / 256, 256, 0, stream>>>(
        convws, fea, gtconf, scale, out_off, out_aff);
}